// DQNNet_25108378812730
// MI455X (gfx1250) — compile-verified
//
#include <hip/hip_runtime.h>
#include <hip/hip_bf16.h>

// ---------------------------------------------------------------------------
// GNN + bilinear readout for MI455X (gfx1250, wave32).
// All dense math runs on V_WMMA_F32_16X16X4_F32 (fp32 WMMA, 16x16x4).
// ---------------------------------------------------------------------------

typedef __attribute__((ext_vector_type(2))) float v2f;
typedef __attribute__((ext_vector_type(8))) float v8f;

#define NN    32768          // nodes
#define HH    64             // hidden
#define EE    524288         // edges
#define GG    256            // graphs
#define PP    128            // nodes per graph

__device__ __forceinline__ v8f wmma4(v2f a, v2f b, v8f c) {
  // D = A(16x4,f32) x B(4x16,f32) + C(16x16,f32)
  return __builtin_amdgcn_wmma_f32_16x16x4_f32(
      false, a, false, b, (short)0, c, false, false);
}

// Fragment loader for fp32 16x16x4 WMMA.
//  A (16x4, MxK): rows rowbase..rowbase+15 of row-major [*, ld] matrix,
//   lane<16  -> row=rowbase+lm, (v0,v1)=(K=k0,  K=k0+1)
//   lane>=16 -> row=rowbase+lm, (v0,v1)=(K=k0+2,K=k0+3)
//  B (4x16, KxN) for y = X @ Wt.T with row-major Wt[out][in]:
//   identical indexing with rowbase = output-column base (B[k][n]=Wt[n][k]).
__device__ __forceinline__ v2f frag(const float* base, int ld, int rowbase, int k0) {
  int lane = threadIdx.x & 31;
  int half = lane >> 4;
  int lm   = lane & 15;
  const float* p = base + (rowbase + lm) * ld + k0 + 2 * half;
  return v2f{p[0], p[1]};
}

// ---------------------------------------------------------------------------
// Fold the depth-layer weights:
//   Wa[o][i] = sum_k W4[o][k]    * W3[k][i]
//   Wb[o][i] = sum_k W4[o][64+k] * W2[k][i]
//   cv[o]    = sum_k W4[o][k]*b3[k] + W4[o][64+k]*b2[k] + b4[o]
// so that h_new = relu(m @ Wa.T + h @ Wb.T + cv), then row-normalize.
// ---------------------------------------------------------------------------
__global__ void __launch_bounds__(256) k_prep(
    const float* __restrict__ W2, const float* __restrict__ b2,
    const float* __restrict__ W3, const float* __restrict__ b3,
    const float* __restrict__ W4, const float* __restrict__ b4,
    float* __restrict__ Wa, float* __restrict__ Wb, float* __restrict__ cv) {
  int t = threadIdx.x;
  for (int e = t * 16; e < t * 16 + 16; ++e) {
    int o = e >> 6, i = e & 63;
    float sa = 0.f, sb = 0.f;
    #pragma unroll 8
    for (int k = 0; k < 64; ++k) {
      sa += W4[o * 128 + k]      * W3[k * 64 + i];
      sb += W4[o * 128 + 64 + k] * W2[k * 64 + i];
    }
    Wa[e] = sa; Wb[e] = sb;
  }
  if (t < 64) {
    float s = b4[t];
    #pragma unroll 8
    for (int k = 0; k < 64; ++k)
      s += W4[t * 128 + k] * b3[k] + W4[t * 128 + 64 + k] * b2[k];
    cv[t] = s;
  }
}

// ---------------------------------------------------------------------------
// h = normalize(relu(x @ W1.T + b1)); one wave per node, 2 outputs per lane.
// ---------------------------------------------------------------------------
__global__ void __launch_bounds__(256) k_embed(
    const float* __restrict__ x, const float* __restrict__ W1,
    const float* __restrict__ b1, float* __restrict__ h) {
  int node = (blockIdx.x * blockDim.x + threadIdx.x) >> 5;
  int lane = threadIdx.x & 31;
  if (node >= NN) return;
  const float* xr = x + node * 8;
  float xv[8];
  #pragma unroll
  for (int j = 0; j < 8; ++j) xv[j] = xr[j];
  float o0 = b1[lane], o1 = b1[lane + 32];
  #pragma unroll
  for (int j = 0; j < 8; ++j) {
    o0 += xv[j] * W1[lane * 8 + j];
    o1 += xv[j] * W1[(lane + 32) * 8 + j];
  }
  o0 = fmaxf(o0, 0.f);
  o1 = fmaxf(o1, 0.f);
  float ss = o0 * o0 + o1 * o1;
  #pragma unroll
  for (int msk = 16; msk >= 1; msk >>= 1) ss += __shfl_xor(ss, msk, 32);
  float inv = rsqrtf(ss);
  h[node * 64 + lane]      = o0 * inv;
  h[node * 64 + lane + 32] = o1 * inv;
}

// ---------------------------------------------------------------------------
// m[dst] += h[src] over all edges. 4 threads/edge, float4 gathers,
// global_atomic_add_f32 scatters.
// ---------------------------------------------------------------------------
__global__ void __launch_bounds__(256) k_scatter(
    const int* __restrict__ src, const int* __restrict__ dst,
    const float* __restrict__ h, float* __restrict__ m) {
  int tid = blockIdx.x * blockDim.x + threadIdx.x;
  if (tid >= EE * 4) return;
  int e = tid >> 2;
  int c = (tid & 3) << 4;
  int s = src[e], d = dst[e];
  const float4* hs = (const float4*)(h + s * 64 + c);
  float* md = m + d * 64 + c;
  #pragma unroll
  for (int q = 0; q < 4; ++q) {
    float4 v = hs[q];
    atomicAdd(md + q * 4 + 0, v.x);
    atomicAdd(md + q * 4 + 1, v.y);
    atomicAdd(md + q * 4 + 2, v.z);
    atomicAdd(md + q * 4 + 3, v.w);
  }
}

// ---------------------------------------------------------------------------
// h_new = rownorm(relu(m @ Wa.T + h @ Wb.T + cv)), in place on h.
// One wave computes a 16(node)x64(out) tile: 4 output 16x16 tiles, K=64 in
// steps of 4, two input matrices => 128 v_wmma_f32_16x16x4_f32 per wave.
// ---------------------------------------------------------------------------
__global__ void __launch_bounds__(256) k_layer(
    const float* __restrict__ m, float* __restrict__ h,
    const float* __restrict__ Wa, const float* __restrict__ Wb,
    const float* __restrict__ cv) {
  __shared__ float sWa[64 * 64];
  __shared__ float sWb[64 * 64];
  __shared__ float sc[64];
  for (int i = threadIdx.x; i < 4096; i += 256) { sWa[i] = Wa[i]; sWb[i] = Wb[i]; }
  if (threadIdx.x < 64) sc[threadIdx.x] = cv[threadIdx.x];
  __syncthreads();

  int wave    = threadIdx.x >> 5;
  int rowbase = (blockIdx.x * 8 + wave) * 16;   // exact: 256 blocks * 8 * 16 = 32768
  int lane = threadIdx.x & 31, half = lane >> 4, lm = lane & 15;

  v8f acc[4] = { {}, {}, {}, {} };
  #pragma unroll
  for (int k0 = 0; k0 < 64; k0 += 4) {
    v2f am = frag(m, 64, rowbase, k0);
    v2f ah = frag(h, 64, rowbase, k0);
    #pragma unroll
    for (int t = 0; t < 4; ++t) {
      acc[t] = wmma4(am, frag(sWa, 64, t * 16, k0), acc[t]);
      acc[t] = wmma4(ah, frag(sWb, 64, t * 16, k0), acc[t]);
    }
  }

  // bias + relu, then per-row L2 norm (rows live in 16-lane halves)
  float vals[4][8];
  float ss[8] = {0, 0, 0, 0, 0, 0, 0, 0};
  #pragma unroll
  for (int t = 0; t < 4; ++t) {
    float cb = sc[t * 16 + lm];
    #pragma unroll
    for (int r = 0; r < 8; ++r) {
      float v = fmaxf(acc[t][r] + cb, 0.f);
      vals[t][r] = v;
      ss[r] += v * v;
    }
  }
  #pragma unroll
  for (int msk = 1; msk <= 8; msk <<= 1)
    #pragma unroll
    for (int r = 0; r < 8; ++r) ss[r] += __shfl_xor(ss[r], msk, 32);

  #pragma unroll
  for (int r = 0; r < 8; ++r) {
    float inv = rsqrtf(ss[r]);
    int row = rowbase + r + 8 * half;
    #pragma unroll
    for (int t = 0; t < 4; ++t)
      h[row * 64 + t * 16 + lm] = vals[t][r] * inv;
  }
}

// ---------------------------------------------------------------------------
// Readout, one block (8 waves) per graph g:
//   M_g[o][i] = sum_j W5[o][i*64+j] * last[j]            (VALU, W5 L2-resident)
//   y1 = others @ M_g.T + b5                             (WMMA, -> LDS)
//   y2 = relu(y1 @ Wd1.T + bd1)                          (WMMA)
//   out = y2 @ Wd2.T + bd2                               (shuffle reduce)
// ---------------------------------------------------------------------------
__global__ void __launch_bounds__(256) k_readout(
    const float* __restrict__ h,  const float* __restrict__ W5,
    const float* __restrict__ b5, const float* __restrict__ Wd1,
    const float* __restrict__ bd1,const float* __restrict__ Wd2,
    const float* __restrict__ bd2,float* __restrict__ out) {
  __shared__ float slast[64];
  __shared__ float sMg[64 * 64];
  __shared__ float sY1[128 * 64];
  int g = blockIdx.x;
  int tid = threadIdx.x;

  if (tid < 64) slast[tid] = h[(g * PP + PP - 1) * 64 + tid];
  __syncthreads();

  // M_g: 4096 entries, 16 per thread, 64-MAC dot each
  for (int e = tid * 16; e < tid * 16 + 16; ++e) {
    int o = e >> 6, i = e & 63;
    const float* wr = W5 + o * 4096 + i * 64;
    float s = 0.f;
    #pragma unroll 8
    for (int j = 0; j < 64; ++j) s += wr[j] * slast[j];
    sMg[o * 64 + i] = s;
  }
  __syncthreads();

  int wave = tid >> 5, lane = tid & 31, half = lane >> 4, lm = lane & 15;
  int prow = wave * 16;                     // graph-local row base (0..112)
  const float* hg = h + g * PP * 64;

  // y1 = rows[prow..prow+15] @ M_g.T + b5  -> LDS
  {
    v8f acc[4] = { {}, {}, {}, {} };
    #pragma unroll
    for (int k0 = 0; k0 < 64; k0 += 4) {
      v2f a = frag(hg, 64, prow, k0);
      #pragma unroll
      for (int t = 0; t < 4; ++t)
        acc[t] = wmma4(a, frag(sMg, 64, t * 16, k0), acc[t]);
    }
    #pragma unroll
    for (int t = 0; t < 4; ++t) {
      float bb = b5[t * 16 + lm];
      #pragma unroll
      for (int r = 0; r < 8; ++r)
        sY1[(prow + r + 8 * half) * 64 + t * 16 + lm] = acc[t][r] + bb;
    }
  }
  __syncthreads();

  // y2 = relu(y1 @ Wd1.T + bd1); fused final dot with Wd2
  v8f acc2[8] = { {}, {}, {}, {}, {}, {}, {}, {} };
  #pragma unroll
  for (int k0 = 0; k0 < 64; k0 += 4) {
    v2f a = frag(sY1, 64, prow, k0);
    #pragma unroll
    for (int t = 0; t < 8; ++t)
      acc2[t] = wmma4(a, frag(Wd1, 64, t * 16, k0), acc2[t]);
  }
  float part[8] = {0, 0, 0, 0, 0, 0, 0, 0};
  #pragma unroll
  for (int t = 0; t < 8; ++t) {
    int col = t * 16 + lm;
    float bb = bd1[col];
    float wv = Wd2[col];
    #pragma unroll
    for (int r = 0; r < 8; ++r)
      part[r] += fmaxf(acc2[t][r] + bb, 0.f) * wv;
  }
  #pragma unroll
  for (int msk = 1; msk <= 8; msk <<= 1)
    #pragma unroll
    for (int r = 0; r < 8; ++r) part[r] += __shfl_xor(part[r], msk, 32);

  if (lm == 0) {
    float bd = bd2[0];
    #pragma unroll
    for (int r = 0; r < 8; ++r) {
      int p = prow + r + 8 * half;
      if (p < PP - 1) out[g * (PP - 1) + p] = part[r] + bd;
    }
  }
}

// ---------------------------------------------------------------------------
extern "C" void kernel_launch(void* const* d_in, const int* in_sizes, int n_in,
                              void* d_out, int out_size, void* d_ws, size_t ws_size,
                              hipStream_t stream) {
  const float* x   = (const float*)d_in[0];
  const int*   ei  = (const int*)d_in[1];   // edge_index (2,E); harness passes int32
  // d_in[2]=batch, d_in[3]=num_graphs: derivable (node/P, G) -> unused
  const float* W1  = (const float*)d_in[4];
  const float* b1  = (const float*)d_in[5];
  const float* W2  = (const float*)d_in[6];
  const float* b2  = (const float*)d_in[7];
  const float* W3  = (const float*)d_in[8];
  const float* b3  = (const float*)d_in[9];
  const float* W4  = (const float*)d_in[10];
  const float* b4  = (const float*)d_in[11];
  const float* W5  = (const float*)d_in[12];
  const float* b5  = (const float*)d_in[13];
  const float* Wd1 = (const float*)d_in[14];
  const float* bd1 = (const float*)d_in[15];
  const float* Wd2 = (const float*)d_in[16];
  const float* bd2 = (const float*)d_in[17];
  float* out = (float*)d_out;

  const int* src = ei;
  const int* dst = ei + EE;

  // workspace: h (8 MB) | m (8 MB) | Wa | Wb | cv   (~16.8 MB total)
  float* ws = (float*)d_ws;
  float* h  = ws;
  float* mB = h  + (size_t)NN * 64;
  float* Wa = mB + (size_t)NN * 64;
  float* Wb = Wa + 4096;
  float* cv = Wb + 4096;

  k_prep<<<1, 256, 0, stream>>>(W2, b2, W3, b3, W4, b4, Wa, Wb, cv);
  k_embed<<<NN / 8, 256, 0, stream>>>(x, W1, b1, h);

  for (int depth = 0; depth < 3; ++depth) {
    hipMemsetAsync(mB, 0, (size_t)NN * 64 * sizeof(float), stream);
    k_scatter<<<(EE * 4) / 256, 256, 0, stream>>>(src, dst, h, mB);
    k_layer<<<NN / 128, 256, 0, stream>>>(mB, h, Wa, Wb, cv);
  }

  k_readout<<<GG, 256, 0, stream>>>(h, W5, b5, Wd1, bd1, Wd2, bd2, out);
}